// SparseGATLayer_21182778704844
// MI455X (gfx1250) — compile-verified
//
#include <hip/hip_runtime.h>
#include <cmath>

typedef __attribute__((ext_vector_type(2))) float v2f;
typedef __attribute__((ext_vector_type(8))) float v8f;

#define N_NODES 100000
#define N_EDGES 1600000
#define DIM     128
#define LRELU_ALPHA 0.2f
#define EPSV    9e-15f

// Monotone float -> uint encoding so we can use integer atomicMax for a global fp max.
__device__ __forceinline__ unsigned encf(float f) {
  unsigned u = __float_as_uint(f);
  return (u & 0x80000000u) ? ~u : (u | 0x80000000u);
}
__device__ __forceinline__ float decf(unsigned k) {
  unsigned u = (k & 0x80000000u) ? (k & 0x7FFFFFFFu) : ~k;
  return __uint_as_float(u);
}

// ---------------------------------------------------------------------------
// 0) Init: zero the atomic accumulators (d_out, rowsum) and the max key.
// ---------------------------------------------------------------------------
__global__ void k_init(float* __restrict__ out, float* __restrict__ rowsum,
                       unsigned* __restrict__ maxkey, int total) {
  int i = blockIdx.x * blockDim.x + threadIdx.x;
  if (i < total)   out[i] = 0.0f;
  if (i < N_NODES) rowsum[i] = 0.0f;
  if (i == 0)      *maxkey = 0u;   // below encoding of every finite float
}

// ---------------------------------------------------------------------------
// 1) h = x @ W via V_WMMA_F32_16X16X4_F32.  One wave owns a 16x16 C tile;
//    8 waves/block cover one 16-row slab across all 128 output columns.
//    K loop: 128/4 = 32 WMMA issues per wave. Exact fp32 math (matches ref).
// ---------------------------------------------------------------------------
__global__ void __launch_bounds__(256) k_gemm(const float* __restrict__ x,
                                              const float* __restrict__ W,
                                              float* __restrict__ h) {
  const int wave = threadIdx.x >> 5;
  const int lane = threadIdx.x & 31;
  const int rc   = lane & 15;   // A: row M; B/C/D: column N
  const int hi   = lane >> 4;   // K-half / M-half selector
  const int m0   = blockIdx.x * 16;
  const int n0   = wave * 16;

  v8f c = {};
  // A frag: lane rc holds A[m0+rc][k0 + 2*hi + {0,1}]  (ISA 16x4 f32 layout)
  const float* xa = x + (size_t)(m0 + rc) * DIM + 2 * hi;
  // B frag: lane rc holds B[k0 + 2*hi + {0,1}][n0+rc]  (row-striped like C/D)
  const float* wb = W + (size_t)(2 * hi) * DIM + n0 + rc;

#pragma unroll
  for (int kk = 0; kk < 32; ++kk) {
    const int k0 = kk * 4;
    v2f a, b;
    a.x = xa[k0 + 0];
    a.y = xa[k0 + 1];
    b.x = wb[(k0 + 0) * DIM];
    b.y = wb[(k0 + 1) * DIM];
    c = __builtin_amdgcn_wmma_f32_16x16x4_f32(false, a, false, b,
                                              (short)0, c, false, false);
  }
#pragma unroll
  for (int r = 0; r < 8; ++r)
    h[(size_t)(m0 + r + 8 * hi) * DIM + n0 + rc] = c[r];
}

// ---------------------------------------------------------------------------
// 2) Per-node attention scores: hs[n] = h[n].a_src, hd[n] = h[n].a_dst.
//    One wave32 per node, float4 per lane, shfl_xor tree reduction.
// ---------------------------------------------------------------------------
__global__ void __launch_bounds__(256) k_scores(const float* __restrict__ h,
                                                const float* __restrict__ attn,
                                                float* __restrict__ hs,
                                                float* __restrict__ hd) {
  const int wave = threadIdx.x >> 5, lane = threadIdx.x & 31;
  const int node = blockIdx.x * 8 + wave;
  if (node >= N_NODES) return;
  const float4 hv = *(const float4*)(h + (size_t)node * DIM + lane * 4);
  const float4 as = *(const float4*)(attn + lane * 4);
  const float4 ad = *(const float4*)(attn + DIM + lane * 4);
  float ps = hv.x * as.x + hv.y * as.y + hv.z * as.z + hv.w * as.w;
  float pd = hv.x * ad.x + hv.y * ad.y + hv.z * ad.z + hv.w * ad.w;
#pragma unroll
  for (int m = 16; m > 0; m >>= 1) {
    ps += __shfl_xor(ps, m, 32);
    pd += __shfl_xor(pd, m, 32);
  }
  if (lane == 0) { hs[node] = ps; hd[node] = pd; }
}

// ---------------------------------------------------------------------------
// 3) Per-edge LeakyReLU logit + global max (block-reduce, 1 atomic / block).
// ---------------------------------------------------------------------------
__global__ void __launch_bounds__(256) k_edge_logit(const long long* __restrict__ ei,
                                                    const float* __restrict__ hs,
                                                    const float* __restrict__ hd,
                                                    float* __restrict__ ea,
                                                    unsigned* __restrict__ maxkey) {
  __shared__ float red[256];
  const int e = blockIdx.x * 256 + threadIdx.x;
  float v = -3.4e38f;
  if (e < N_EDGES) {
    const int s = (int)ei[e];
    const int d = (int)ei[N_EDGES + e];
    const float val = hs[s] + hd[d];
    v = val > 0.0f ? val : LRELU_ALPHA * val;
    ea[e] = v;
  }
  red[threadIdx.x] = v;
  __syncthreads();
#pragma unroll
  for (int st = 128; st > 0; st >>= 1) {
    if (threadIdx.x < st)
      red[threadIdx.x] = fmaxf(red[threadIdx.x], red[threadIdx.x + st]);
    __syncthreads();
  }
  if (threadIdx.x == 0) atomicMax(maxkey, encf(red[0]));
}

// ---------------------------------------------------------------------------
// 4) Edge scatter: w = exp(ea - max); rowsum[src] += w; out[src] += w*h[dst].
//    One wave32 per edge; 4 floats/lane; native global_atomic_add_f32.
// ---------------------------------------------------------------------------
__global__ void __launch_bounds__(256) k_scatter(const long long* __restrict__ ei,
                                                 const float* __restrict__ ea,
                                                 const float* __restrict__ h,
                                                 const unsigned* __restrict__ maxkey,
                                                 float* __restrict__ rowsum,
                                                 float* __restrict__ out) {
  const int wave = threadIdx.x >> 5, lane = threadIdx.x & 31;
  const int e = blockIdx.x * 8 + wave;
  if (e >= N_EDGES) return;
  const float maxv = decf(*maxkey);
  const int s = (int)ei[e];
  const int d = (int)ei[N_EDGES + e];
  const float w = expf(ea[e] - maxv);
  if (lane == 0) unsafeAtomicAdd(&rowsum[s], w);
  const float4 hv = *(const float4*)(h + (size_t)d * DIM + lane * 4);
  float* ob = out + (size_t)s * DIM + lane * 4;
  unsafeAtomicAdd(ob + 0, w * hv.x);
  unsafeAtomicAdd(ob + 1, w * hv.y);
  unsafeAtomicAdd(ob + 2, w * hv.z);
  unsafeAtomicAdd(ob + 3, w * hv.w);
}

// ---------------------------------------------------------------------------
// 5) Finalize in place: out = elu(out / (rowsum + EPS))  (elu alpha = 1).
// ---------------------------------------------------------------------------
__global__ void k_final(float* __restrict__ out, const float* __restrict__ rowsum,
                        int total) {
  const int i = blockIdx.x * 256 + threadIdx.x;
  if (i >= total) return;
  const int n = i >> 7;  // /DIM
  const float v = out[i] / (rowsum[n] + EPSV);
  out[i] = v > 0.0f ? v : expm1f(v);
}

// ---------------------------------------------------------------------------
extern "C" void kernel_launch(void* const* d_in, const int* in_sizes, int n_in,
                              void* d_out, int out_size, void* d_ws, size_t ws_size,
                              hipStream_t stream) {
  const float*     x    = (const float*)d_in[0];
  const long long* ei   = (const long long*)d_in[1];  // int64 edge_index (2,E)
  const float*     W    = (const float*)d_in[2];
  const float*     attn = (const float*)d_in[3];
  float* out = (float*)d_out;

  // Workspace carving (floats): h | hs | hd | ea | rowsum | maxkey  (~59 MB)
  float* wsf = (float*)d_ws;
  size_t off = 0;
  float* h      = wsf + off; off += (size_t)N_NODES * DIM;
  float* hs     = wsf + off; off += N_NODES;
  float* hd     = wsf + off; off += N_NODES;
  float* ea     = wsf + off; off += N_EDGES;
  float* rowsum = wsf + off; off += N_NODES;
  unsigned* maxkey = (unsigned*)(wsf + off);

  const int total = N_NODES * DIM;  // 12,800,000 == out_size

  k_init<<<(total + 255) / 256, 256, 0, stream>>>(out, rowsum, maxkey, total);
  k_gemm<<<N_NODES / 16, 256, 0, stream>>>(x, W, h);              // 6250 blocks
  k_scores<<<(N_NODES + 7) / 8, 256, 0, stream>>>(h, attn, hs, hd);
  k_edge_logit<<<(N_EDGES + 255) / 256, 256, 0, stream>>>(ei, hs, hd, ea, maxkey);
  k_scatter<<<(N_EDGES + 7) / 8, 256, 0, stream>>>(ei, ea, h, maxkey, rowsum, out);
  k_final<<<(total + 255) / 256, 256, 0, stream>>>(out, rowsum, total);
}